// RandomVectorQuantizer_12481174962926
// MI455X (gfx1250) — compile-verified
//
#include <hip/hip_runtime.h>
#include <hip/hip_bf16.h>
#include <math.h>
#include <stdint.h>

typedef __attribute__((ext_vector_type(2))) float v2f;
typedef __attribute__((ext_vector_type(8))) float v8f;

#define BT     16384   // B*T
#define IN_DIM 1024
#define GD     512     // G*D
#define NG     8
#define DD     64
#define NV     2048

// ---------------------------------------------------------------------------
// CDNA5 async global->LDS copy (ASYNCcnt path), per CDNA5 ISA.
// Each call copies 16 bytes per active lane from the global address gaddr to the LDS address lds_addr.
// ---------------------------------------------------------------------------
__device__ __forceinline__ void async_ld_b128(unsigned lds_addr, const void* gaddr) {
  asm volatile("global_load_async_to_lds_b128 %0, %1, off"
               :: "v"(lds_addr), "v"((unsigned long long)(uintptr_t)gaddr)
               : "memory");
}
__device__ __forceinline__ void wait_async0() {
  asm volatile("s_wait_asynccnt 0x0" ::: "memory");
}

// ---------------------------------------------------------------------------
// Kernel 1: x = inputs(16384x1024) @ proj(1024x512), fp32 WMMA 16x16x4.
// Block = 8 waves sharing one 16-row A tile. A is staged through LDS with
// double-buffered async DMA (global_load_async_to_lds_b128), overlapping the
// next chunk's copy with the current chunk's 16 WMMAs.
// Output written to xn (== quantized region of d_out, reused as scratch).
// ---------------------------------------------------------------------------
__global__ __launch_bounds__(256)
void k_proj_gemm(const float* __restrict__ inp, const float* __restrict__ proj,
                 float* __restrict__ xn) {
  __shared__ float ldsA[2][16 * 64];                 // 2 x 4KB A chunks

  const int t    = threadIdx.x;
  const int lane = t & 31;
  const int wave = t >> 5;
  const int tm   = blockIdx.x >> 2;                  // M tile: 0..1023
  const int tn   = ((blockIdx.x & 3) << 3) + wave;   // N tile: 0..31
  const int row0 = tm << 4;
  const int col0 = tn << 4;
  const int nl   = lane & 15;                        // A row (m) == B col (n)
  const int half = lane >> 4;                        // K pair select

  // Per-thread 16B slice of the 16x64 A chunk (4KB / 256 threads)
  const int ar   = t >> 4;                           // chunk row 0..15
  const int ac   = (t & 15) * 4;                     // chunk col 0..60
  const float* asrc = inp + (size_t)(row0 + ar) * IN_DIM + ac;
  const unsigned lds0 = (unsigned)(uintptr_t)&ldsA[0][0];
  const unsigned ldst = lds0 + (unsigned)((ar * 64 + ac) * sizeof(float));
  const unsigned bufbytes = 16 * 64 * sizeof(float); // 4096

  async_ld_b128(ldst, asrc);                         // chunk 0 -> buf 0

  const float* bcol = proj + col0 + nl;
  v8f acc = {};
  for (int c = 0; c < IN_DIM / 64; ++c) {            // 16 K-chunks of 64
    const int cur = c & 1;
    wait_async0();
    __syncthreads();                                 // buf[cur] visible to all
    if (c + 1 < IN_DIM / 64)
      async_ld_b128(ldst + (unsigned)((c + 1) & 1) * bufbytes,
                    asrc + (size_t)(c + 1) * 64);    // prefetch next chunk

    const float* abuf = &ldsA[cur][0];
    const float* bp0  = bcol + (size_t)(c * 64 + half * 2) * GD;
#pragma unroll
    for (int kk = 0; kk < 16; ++kk) {
      v2f a = *(const v2f*)(abuf + nl * 64 + kk * 4 + half * 2);
      v2f b;
      b.x = bp0[(size_t)(kk * 4) * GD];              // B[k  ][n]
      b.y = bp0[(size_t)(kk * 4) * GD + GD];         // B[k+1][n]
      acc = __builtin_amdgcn_wmma_f32_16x16x4_f32(false, a, false, b,
                                                  (short)0, acc, false, false);
    }
    __syncthreads();                                 // done reading buf[cur]
  }

  // D layout: lane&15 = col, VGPR r = row (r or r+8 by half)
  float* dst = xn + (size_t)(row0 + half * 8) * GD + col0 + nl;
#pragma unroll
  for (int r = 0; r < 8; ++r)
    dst[(size_t)r * GD] = acc[r];
}

// ---------------------------------------------------------------------------
// Kernel 1b: in-place L2 normalize each (token,group) 64-vector.
// One wave per vector; wave32 __shfl_xor reduction.
// ---------------------------------------------------------------------------
__global__ __launch_bounds__(256)
void k_l2norm(float* __restrict__ xn) {
  const int lane = threadIdx.x & 31;
  const int wave = threadIdx.x >> 5;
  const int pair = blockIdx.x * 8 + wave;            // token*8 + g
  float* p = xn + (size_t)pair * DD + lane * 2;
  v2f v = *(const v2f*)p;
  float ss = v.x * v.x + v.y * v.y;
#pragma unroll
  for (int s = 16; s > 0; s >>= 1)
    ss += __shfl_xor(ss, s, 32);
  const float scale = 1.0f / fmaxf(sqrtf(ss), 1e-12f);
  v.x *= scale; v.y *= scale;
  *(v2f*)p = v;
}

// ---------------------------------------------------------------------------
// Kernel 2: sim = xn . codebook^T per group, fused argmax over V.
// Block = 16 tokens x 8 waves (one wave per group). A slice cached in regs,
// codebook streamed in 16-wide N tiles (contiguous float2 loads + prefetch).
// Argmax with first-index tie-break via cross-lane xor shuffles.
// ---------------------------------------------------------------------------
__global__ __launch_bounds__(256)
void k_sim_argmax(const float* __restrict__ xn, const float* __restrict__ cb,
                  float* __restrict__ ids_out) {
  const int lane = threadIdx.x & 31;
  const int g    = threadIdx.x >> 5;                 // wave id == group
  const int tok0 = blockIdx.x << 4;
  const int nl   = lane & 15;
  const int half = lane >> 4;

  // Preload A: 16 K-steps of 16x4 (this wave's 16 tokens x 64 dims)
  v2f a[16];
  const float* ap = xn + (size_t)(tok0 + nl) * GD + g * DD + half * 2;
#pragma unroll
  for (int kk = 0; kk < 16; ++kk)
    a[kk] = *(const v2f*)(ap + kk * 4);

  float best_v[8];
  int   best_i[8];
#pragma unroll
  for (int r = 0; r < 8; ++r) { best_v[r] = -3.402823466e+38f; best_i[r] = 0; }

  // B[k][n] = codebook[v0+n][g][k]; k contiguous in memory -> float2 loads
  const float* bbase = cb + (size_t)nl * GD + g * DD + half * 2;
  for (int nt = 0; nt < NV / 16; ++nt) {
    const float* bp = bbase + (size_t)nt * 16 * GD;
    __builtin_prefetch(bp + 16 * GD, 0, 0);          // global_prefetch_b8
    v8f acc = {};
#pragma unroll
    for (int kk = 0; kk < 16; ++kk) {
      v2f b = *(const v2f*)(bp + kk * 4);
      acc = __builtin_amdgcn_wmma_f32_16x16x4_f32(false, a[kk], false, b,
                                                  (short)0, acc, false, false);
    }
    const int idbase = nt * 16 + nl;
#pragma unroll
    for (int r = 0; r < 8; ++r) {
      float v  = acc[r];
      int   id = idbase;
#pragma unroll
      for (int s = 1; s < 16; s <<= 1) {             // reduce over columns
        float ov = __shfl_xor(v, s, 32);
        int   oi = __shfl_xor(id, s, 32);
        if (ov > v || (ov == v && oi < id)) { v = ov; id = oi; }
      }
      if (v > best_v[r] || (v == best_v[r] && id < best_i[r])) {
        best_v[r] = v; best_i[r] = id;
      }
    }
  }
  // lane 0 holds rows 0..7, lane 16 holds rows 8..15
  if (nl == 0) {
#pragma unroll
    for (int r = 0; r < 8; ++r)
      ids_out[(size_t)(tok0 + half * 8 + r) * NG + g] = (float)best_i[r];
  }
}

// ---------------------------------------------------------------------------
// Kernel 3: quantized = codebook[id] * mask (overwrites xn scratch region),
// ids -> -1 where padded, histogram + frame-count atomics.
// ---------------------------------------------------------------------------
__global__ __launch_bounds__(256)
void k_gather(float* __restrict__ ids_out, const float* __restrict__ paddings,
              const float* __restrict__ cb, float* __restrict__ quant,
              unsigned* __restrict__ hist, float* __restrict__ frames) {
  const int lane  = threadIdx.x & 31;
  const int wave  = threadIdx.x >> 5;
  const int pair  = blockIdx.x * 8 + wave;           // token*8 + g
  const int token = pair >> 3;
  const int g     = pair & 7;

  const int   id   = (int)ids_out[pair];
  const float pad  = paddings[token];
  const float mask = 1.0f - pad;

  v2f q = *(const v2f*)(cb + (size_t)id * GD + g * DD + lane * 2);
  q.x *= mask; q.y *= mask;
  *(v2f*)(quant + (size_t)pair * DD + lane * 2) = q;

  if (lane == 0) {
    if (pad > 0.0f)  ids_out[pair] = -1.0f;
    if (mask > 0.0f) atomicAdd(&hist[g * NV + id], 1u);
    if (g == 0)      atomicAdd(frames, mask);
  }
}

// ---------------------------------------------------------------------------
// Kernel 4: per-group entropy -> pplx (mean of exp) and coverage.
// One block per group.
// ---------------------------------------------------------------------------
__global__ __launch_bounds__(256)
void k_stats(const unsigned* __restrict__ hist, const float* __restrict__ frames,
             float* __restrict__ pplx, float* __restrict__ cov) {
  __shared__ float se[256];
  __shared__ float sc[256];
  const int g = blockIdx.x;
  const float nf = fmaxf(frames[0], 1.0f);
  float e = 0.0f, c = 0.0f;
  for (int v = threadIdx.x; v < NV; v += 256) {
    const float p = (float)hist[g * NV + v] / nf;
    e -= p * logf(p + 1e-10f);
    if (p > 0.0f) c += 1.0f;
  }
  se[threadIdx.x] = e; sc[threadIdx.x] = c;
  __syncthreads();
  for (int s = 128; s > 0; s >>= 1) {
    if (threadIdx.x < s) {
      se[threadIdx.x] += se[threadIdx.x + s];
      sc[threadIdx.x] += sc[threadIdx.x + s];
    }
    __syncthreads();
  }
  if (threadIdx.x == 0) {
    atomicAdd(pplx, expf(se[0]) / (float)NG);
    atomicAdd(cov,  sc[0] / (float)(NG * NV));
  }
}

// ---------------------------------------------------------------------------
extern "C" void kernel_launch(void* const* d_in, const int* in_sizes, int n_in,
                              void* d_out, int out_size, void* d_ws, size_t ws_size,
                              hipStream_t stream) {
  (void)in_sizes; (void)n_in; (void)out_size; (void)ws_size;
  const float* inputs   = (const float*)d_in[0];  // (8,2048,1024)
  const float* paddings = (const float*)d_in[1];  // (8,2048)
  const float* proj     = (const float*)d_in[2];  // (1024,512)
  const float* codebook = (const float*)d_in[3];  // (2048,8,64)

  float* out      = (float*)d_out;
  float* ids_out  = out;                              // 16384*8 floats
  float* quant    = out + (size_t)BT * NG;            // 16384*512 floats (also xn scratch)
  float* pplx     = out + (size_t)BT * NG + (size_t)BT * GD;
  float* cov      = pplx + 1;

  unsigned* hist   = (unsigned*)d_ws;                 // 8*2048 u32
  float*    frames = (float*)((char*)d_ws + (size_t)NG * NV * sizeof(unsigned));

  hipMemsetAsync(d_ws, 0, (size_t)NG * NV * sizeof(unsigned) + 16, stream);
  hipMemsetAsync(pplx, 0, 2 * sizeof(float), stream);

  // 1) projection GEMM (x stored into quant region as scratch)
  k_proj_gemm<<<dim3((BT / 16) * 4), dim3(256), 0, stream>>>(inputs, proj, quant);
  // 2) per-(token,group) L2 normalize in place
  k_l2norm<<<dim3(BT * NG / 8), dim3(256), 0, stream>>>(quant);
  // 3) similarity + fused argmax
  k_sim_argmax<<<dim3(BT / 16), dim3(256), 0, stream>>>(quant, codebook, ids_out);
  // 4) gather/mask/ids/histogram (overwrites scratch with quantized output)
  k_gather<<<dim3(BT * NG / 8), dim3(256), 0, stream>>>(ids_out, paddings, codebook,
                                                        quant, hist, frames);
  // 5) entropy / pplx / coverage
  k_stats<<<dim3(NG), dim3(256), 0, stream>>>(hist, frames, pplx, cov);
}